// GVAE_21904333210069
// MI455X (gfx1250) — compile-verified
//
#include <hip/hip_runtime.h>
#include <math.h>

#define HEADS 3
#define HD 60
#define QKV (HEADS * HD)   /* 180 */
#define NG 512
#define CDIV(a, b) (((a) + (b) - 1) / (b))

typedef __attribute__((ext_vector_type(2))) float v2f;
typedef __attribute__((ext_vector_type(8))) float v8f;

#define WMMA_F32(a, b, c) \
  __builtin_amdgcn_wmma_f32_16x16x4_f32(false, (a), false, (b), (short)0, (c), false, false)

#if __has_builtin(__builtin_amdgcn_global_load_async_to_lds_b32) && \
    __has_builtin(__builtin_amdgcn_s_wait_asynccnt)
#define GVAE_ASYNC_LDS 1
typedef __attribute__((address_space(1))) int* gvae_gas_ip;   // global (AS1)
typedef __attribute__((address_space(3))) int* gvae_las_ip;   // LDS (AS3)
#else
#define GVAE_ASYNC_LDS 0
#endif

// ---------------------------------------------------------------------------
// fp32 WMMA GEMM:  C[M,N] = A[M,K] @ W[K,N] + bias[N]  (optional ReLU)
// Block = 256 threads = 8 waves computes a 128(M) x 64(N) tile.
// B is staged through LDS once per block in exact WMMA-fragment layout:
//   per k-step s (4 rows) / sub-tile j / lane l: the pair
//   (W[k0+4s+2h][c], W[k0+4s+2h+1][c]) with h=l>>4, c=j*16+(l&15)
//   sits at float index ((s*4+j)*32 + l)*2  ->  one ds_load_b64 per fragment
//   (compiler fuses pairs into ds_load_2addr_b64).
// Staging uses GLOBAL_LOAD_ASYNC_TO_LDS_B32 (cache -> LDS direct, ASYNCcnt)
// when the toolchain exposes it; otherwise a VGPR round-trip.
// Out-of-range k rows are zero-filled at staging (handles K=1369 tail);
// out-of-range B columns are clamped (garbage lands in never-stored D cols).
// M must be a multiple of 16 (50000, 512: both are).
// ---------------------------------------------------------------------------
__global__ __launch_bounds__(256) void gvae_gemm_wmma(
    const float* __restrict__ A, const float* __restrict__ W,
    const float* __restrict__ bias, float* __restrict__ C,
    int M, int K, int N, int relu)
{
  __shared__ __align__(16) float bsh[2048];     // 32 k-rows x 64 cols, 8 KB
  const int wave = threadIdx.x >> 5;
  const int lane = threadIdx.x & 31;
  const int half = lane >> 4;
  const int lm   = lane & 15;
  const int tilesM = M >> 4;
  const int tn  = blockIdx.x;
  const int tmg = blockIdx.y * 8 + wave;
  const bool storeOK = tmg < tilesM;            // excess waves still barrier
  const int tm = storeOK ? tmg : tilesM - 1;

  const float* __restrict__ Abase = A + (size_t)(tm * 16 + lm) * K;
  const float* __restrict__ Ap    = Abase + half * 2;
  const v2f*   __restrict__ bfrag = (const v2f*)bsh;

  const int cb0 = tn * 64 + lm;

  v8f acc0 = {}, acc1 = {}, acc2 = {}, acc3 = {};

  const int fullCh = K >> 5;                    // chunks with all 32 k valid
  const int nCh    = (K + 31) >> 5;

  for (int ch = 0; ch < nCh; ++ch) {
    const int k0 = ch << 5;
    // ---- cooperative stage of B chunk (coalesced, pre-swizzled) ----
    #pragma unroll
    for (int i = 0; i < 8; ++i) {
      int id = threadIdx.x + (i << 8);          // 0..2047
      int r = id >> 6, c = id & 63;
      int krow = k0 + r;
      int cc = tn * 64 + c; cc = cc < N ? cc : N - 1;
      int s = r >> 2, h = (r >> 1) & 1, par = r & 1, j = c >> 4;
      int ln = h * 16 + (c & 15);
      int lidx = ((((s << 2) + j) << 5) + ln) * 2 + par;
#if GVAE_ASYNC_LDS
      if (krow < K) {                           // EXEC-masked async copy
        __builtin_amdgcn_global_load_async_to_lds_b32(
            (gvae_gas_ip)(W + (size_t)krow * N + cc),
            (gvae_las_ip)&bsh[lidx], 0, 0);
      } else {
        bsh[lidx] = 0.f;                        // zero-fill k tail
      }
#else
      int kc = krow < K ? krow : K - 1;
      float v = W[(size_t)kc * N + cc];
      v = (krow < K) ? v : 0.f;                 // zero-fill k tail
      bsh[lidx] = v;
#endif
    }
#if GVAE_ASYNC_LDS
    __builtin_amdgcn_s_wait_asynccnt(0);        // own async copies done
#endif
    __syncthreads();

    if (ch < fullCh) {                          // branch uniform per block
      #pragma unroll
      for (int s = 0; s < 8; ++s) {
        v2f a;  a.x = Ap[s * 4];  a.y = Ap[s * 4 + 1];
        v2f b0 = bfrag[((s << 2) + 0) * 32 + lane];
        v2f b1 = bfrag[((s << 2) + 1) * 32 + lane];
        v2f b2 = bfrag[((s << 2) + 2) * 32 + lane];
        v2f b3 = bfrag[((s << 2) + 3) * 32 + lane];
        acc0 = WMMA_F32(a, b0, acc0);
        acc1 = WMMA_F32(a, b1, acc1);
        acc2 = WMMA_F32(a, b2, acc2);
        acc3 = WMMA_F32(a, b3, acc3);
      }
      Ap += 32;
    } else {                                    // tail chunk: mask A too
      #pragma unroll
      for (int s = 0; s < 8; ++s) {
        const int ka = k0 + s * 4 + half * 2;
        const int kc0 = ka < K ? ka : K - 1;    // clamped in-row, safe
        const int kc1 = ka + 1 < K ? ka + 1 : K - 1;
        float t0 = Abase[kc0], t1 = Abase[kc1];
        v2f a;
        a.x = (ka < K)     ? t0 : 0.f;
        a.y = (ka + 1 < K) ? t1 : 0.f;
        v2f b0 = bfrag[((s << 2) + 0) * 32 + lane];
        v2f b1 = bfrag[((s << 2) + 1) * 32 + lane];
        v2f b2 = bfrag[((s << 2) + 2) * 32 + lane];
        v2f b3 = bfrag[((s << 2) + 3) * 32 + lane];
        acc0 = WMMA_F32(a, b0, acc0);
        acc1 = WMMA_F32(a, b1, acc1);
        acc2 = WMMA_F32(a, b2, acc2);
        acc3 = WMMA_F32(a, b3, acc3);
      }
    }
    __syncthreads();
  }

  // ---- store (post-WMMA; divergence fine) ----
  if (storeOK) {
    const int rowBase = tm * 16 + half * 8;
    auto store_tile = [&](const v8f& ac, int cb) {
      if (cb < N) {
        const float bv = bias[cb];
        #pragma unroll
        for (int r = 0; r < 8; ++r) {
          float val = ac[r] + bv;
          if (relu) val = fmaxf(val, 0.f);
          C[(size_t)(rowBase + r) * N + cb] = val;
        }
      }
    };
    store_tile(acc0, cb0);
    store_tile(acc1, cb0 + 16);
    store_tile(acc2, cb0 + 32);
    store_tile(acc3, cb0 + 48);
  }
}

// ---------------------------------------------------------------------------
__global__ void gvae_fill(float* __restrict__ p, float v, int n) {
  int t = blockIdx.x * blockDim.x + threadIdx.x;
  if (t < n) p[t] = v;
}

__global__ void gvae_relu(float* __restrict__ p, int n) {
  int t = blockIdx.x * blockDim.x + threadIdx.x;
  if (t < n) p[t] = fmaxf(p[t], 0.f);
}

__device__ inline void atomic_max_f32(float* addr, float val) {
  // signed-int max for non-negative floats, unsigned-int min for negatives;
  // consistent with -inf init.
  if (val >= 0.f) atomicMax((int*)addr, __float_as_int(val));
  else            atomicMin((unsigned int*)addr, __float_as_uint(val));
}

// score[e,h] = <q[dst], k[src]> / sqrt(60); also per-(dst,head) running max
__global__ void gvae_edge_scores(const float* __restrict__ q,
                                 const float* __restrict__ k,
                                 const int* __restrict__ src,
                                 const int* __restrict__ dst,
                                 float* __restrict__ score,
                                 float* __restrict__ mbuf, int E)
{
  int t = blockIdx.x * blockDim.x + threadIdx.x;
  if (t >= E * HEADS) return;
  int e = t / HEADS, h = t - e * HEADS;
  int s = src[e], d = dst[e];
  const float* __restrict__ qp = q + (size_t)d * QKV + h * HD;
  const float* __restrict__ kp = k + (size_t)s * QKV + h * HD;
  float acc = 0.f;
  #pragma unroll
  for (int c = 0; c < HD; ++c) acc = fmaf(qp[c], kp[c], acc);
  acc *= 0.12909944487358055994f;   // 1/sqrt(60)
  score[t] = acc;
  atomic_max_f32(&mbuf[d * HEADS + h], acc);
}

// score[e,h] <- exp(score - m[dst]); denom[dst,h] += score
__global__ void gvae_edge_expsum(const int* __restrict__ dst,
                                 const float* __restrict__ mbuf,
                                 float* __restrict__ score,
                                 float* __restrict__ denom, int E)
{
  int t = blockIdx.x * blockDim.x + threadIdx.x;
  if (t >= E * HEADS) return;
  int e = t / HEADS, h = t - e * HEADS;
  int d = dst[e];
  float ex = expf(score[t] - mbuf[d * HEADS + h]);
  score[t] = ex;
  atomicAdd(&denom[d * HEADS + h], ex);
}

// out[dst,c] += (1/3) * sum_h alpha[e,h] * v[src,h,c]   (head-mean folded in)
__global__ void gvae_edge_scatter(const int* __restrict__ src,
                                  const int* __restrict__ dst,
                                  const float* __restrict__ score,
                                  const float* __restrict__ denom,
                                  const float* __restrict__ v,
                                  float* __restrict__ out, int E)
{
  int t = blockIdx.x * blockDim.x + threadIdx.x;
  if (t >= E * HD) return;
  int e = t / HD, c = t - e * HD;
  int s = src[e], d = dst[e];
  float acc = 0.f;
  #pragma unroll
  for (int h = 0; h < HEADS; ++h) {
    float alpha = score[e * HEADS + h] / (denom[d * HEADS + h] + 1e-16f);
    acc = fmaf(alpha, v[(size_t)s * QKV + h * HD + c], acc);
  }
  atomicAdd(&out[(size_t)d * HD + c], acc * (1.f / 3.f));
}

// Per-column biased mean/var over M rows (one block per column)
__global__ __launch_bounds__(256) void gvae_bn_stats(
    const float* __restrict__ X, float* __restrict__ mu,
    float* __restrict__ rstd, int M, int C)
{
  __shared__ float ss[256], sq[256];
  const int c = blockIdx.x;
  float s = 0.f, q2 = 0.f;
  for (int r = threadIdx.x; r < M; r += 256) {
    float x = X[(size_t)r * C + c];
    s += x; q2 = fmaf(x, x, q2);
  }
  ss[threadIdx.x] = s; sq[threadIdx.x] = q2;
  __syncthreads();
  for (int st = 128; st > 0; st >>= 1) {
    if (threadIdx.x < st) {
      ss[threadIdx.x] += ss[threadIdx.x + st];
      sq[threadIdx.x] += sq[threadIdx.x + st];
    }
    __syncthreads();
  }
  if (threadIdx.x == 0) {
    float m = ss[0] / (float)M;
    float var = sq[0] / (float)M - m * m;     // biased (training-mode BN)
    mu[c] = m;
    rstd[c] = rsqrtf(var + 1e-5f);
  }
}

__global__ void gvae_bn_apply(float* __restrict__ X,
                              const float* __restrict__ mu,
                              const float* __restrict__ rstd,
                              const float* __restrict__ g,
                              const float* __restrict__ b, int n, int C)
{
  int t = blockIdx.x * blockDim.x + threadIdx.x;
  if (t >= n) return;
  int c = t % C;
  X[t] = g[c] * (X[t] - mu[c]) * rstd[c] + b[c];
}

__global__ void gvae_pool_accum(const float* __restrict__ emb,
                                const int* __restrict__ batch,
                                float* __restrict__ sums,
                                float* __restrict__ cnt, int Nn)
{
  int t = blockIdx.x * blockDim.x + threadIdx.x;
  if (t >= Nn * HD) return;
  int n = t / HD, c = t - n * HD;
  int b = batch[n];
  atomicAdd(&sums[b * HD + c], emb[t]);
  if (c == 0) atomicAdd(&cnt[b], 1.f);
}

__global__ void gvae_pool_final(const float* __restrict__ sums,
                                const float* __restrict__ cnt,
                                float* __restrict__ z,
                                float* __restrict__ zcat)
{
  int t = blockIdx.x * blockDim.x + threadIdx.x;
  if (t >= NG * HD) return;
  int g = t / HD, c = t - g * HD;
  float val = sums[t] / fmaxf(cnt[g], 1.f);
  z[t] = val;
  zcat[g * 2 * HD + c] = val;         // z duplicated: concat([z, z])
  zcat[g * 2 * HD + HD + c] = val;
}

// ---------------------------------------------------------------------------
extern "C" void kernel_launch(void* const* d_in, const int* in_sizes, int n_in,
                              void* d_out, int out_size, void* d_ws, size_t ws_size,
                              hipStream_t stream)
{
  (void)n_in; (void)out_size; (void)ws_size;
  const float* x     = (const float*)d_in[0];
  const int*   ei    = (const int*)d_in[1];
  const int*   batch = (const int*)d_in[2];
  const int E  = in_sizes[1] / 2;      // 400000
  const int Nn = in_sizes[2];          // 50000 (multiple of 16)
  const int* src = ei;                 // edge_index[0]
  const int* dst = ei + E;             // edge_index[1]
  auto p = [&](int i) { return (const float*)d_in[i]; };

  // ---- workspace carve-up (floats) ----
  float* ws = (float*)d_ws;
  size_t o = 0;
  float* qb    = ws + o; o += (size_t)Nn * QKV;
  float* kb    = ws + o; o += (size_t)Nn * QKV;
  float* vb    = ws + o; o += (size_t)Nn * QKV;
  float* hA    = ws + o; o += (size_t)Nn * HD;
  float* hB    = ws + o; o += (size_t)Nn * HD;
  float* mb    = ws + o; o += (size_t)Nn * HEADS;
  float* db    = ws + o; o += (size_t)Nn * HEADS;
  float* sc    = ws + o; o += (size_t)E * HEADS;
  float* pools = ws + o; o += (size_t)NG * HD;
  float* cnt   = ws + o; o += NG;
  float* zbuf  = ws + o; o += (size_t)NG * HD;
  float* zcat  = ws + o; o += (size_t)NG * 2 * HD;
  float* mA    = ws + o; o += (size_t)NG * 1376;
  float* mB    = ws + o; o += (size_t)NG * 1376;
  float* mu    = ws + o; o += 1376;
  float* rstd  = ws + o; o += 1376;

  auto gemm = [&](const float* A, const float* W, const float* bias, float* C,
                  int M, int K, int N, int relu) {
    dim3 grid((N + 63) >> 6, CDIV(M >> 4, 8));
    gvae_gemm_wmma<<<grid, 256, 0, stream>>>(A, W, bias, C, M, K, N, relu);
  };
  auto fill = [&](float* ptr, float v, int n) {
    gvae_fill<<<CDIV(n, 256), 256, 0, stream>>>(ptr, v, n);
  };
  auto run_bn = [&](float* X, int M, int C, const float* g, const float* b) {
    gvae_bn_stats<<<C, 256, 0, stream>>>(X, mu, rstd, M, C);
    gvae_bn_apply<<<CDIV(M * C, 256), 256, 0, stream>>>(X, mu, rstd, g, b, M * C, C);
  };
  auto run_conv = [&](const float* in, int K, int pw, float* out) {
    // pw = first param index of this conv's 8 params (Wq,bq,Wk,bk,Wv,bv,Ws,bs)
    gemm(in, p(pw + 0), p(pw + 1), qb, Nn, K, QKV, 0);
    gemm(in, p(pw + 2), p(pw + 3), kb, Nn, K, QKV, 0);
    gemm(in, p(pw + 4), p(pw + 5), vb, Nn, K, QKV, 0);
    gemm(in, p(pw + 6), p(pw + 7), out, Nn, K, HD, 0);    // skip connection
    fill(mb, -INFINITY, Nn * HEADS);
    fill(db, 0.f, Nn * HEADS);
    gvae_edge_scores <<<CDIV(E * HEADS, 256), 256, 0, stream>>>(qb, kb, src, dst, sc, mb, E);
    gvae_edge_expsum <<<CDIV(E * HEADS, 256), 256, 0, stream>>>(dst, mb, sc, db, E);
    gvae_edge_scatter<<<CDIV(E * HD,    256), 256, 0, stream>>>(src, dst, sc, db, vb, out, E);
    gvae_relu<<<CDIV(Nn * HD, 256), 256, 0, stream>>>(out, Nn * HD);
  };
  auto mlp_layer = [&](const float* in, int K, int N, int pw, float* out, bool final_) {
    gemm(in, p(pw), p(pw + 1), out, NG, K, N, final_ ? 0 : 1);
    if (!final_) run_bn(out, NG, N, p(pw + 2), p(pw + 3));
  };

  // ---- GNN encoder ----
  run_conv(x,  128, 3, hA);  run_bn(hA, Nn, HD, p(27), p(28));   // conv1 + bn1
  run_conv(hA, HD, 11, hB);  run_bn(hB, Nn, HD, p(29), p(30));   // conv2 + bn2
  run_conv(hB, HD, 19, hA);                                      // zeta (relu only)

  // ---- global mean pool ----
  fill(pools, 0.f, NG * HD);
  fill(cnt,   0.f, NG);
  gvae_pool_accum<<<CDIV(Nn * HD, 256), 256, 0, stream>>>(hA, batch, pools, cnt, Nn);
  gvae_pool_final<<<CDIV(NG * HD, 256), 256, 0, stream>>>(pools, cnt, zbuf, zcat);

  // ---- decoder MLPs: d_out = [xatt | xadj] ----
  float* xatt = (float*)d_out;
  float* xadj = (float*)d_out + (size_t)NG * 1369;

  // adj: 60 -> 256 -> 768 -> 768 -> 1024 -> 1369 -> 1369
  mlp_layer(zbuf,   60,  256, 31, mA, false);
  mlp_layer(mA,    256,  768, 35, mB, false);
  mlp_layer(mB,    768,  768, 39, mA, false);
  mlp_layer(mA,    768, 1024, 43, mB, false);
  mlp_layer(mB,   1024, 1369, 47, mA, false);
  mlp_layer(mA,   1369, 1369, 51, xadj, true);

  // att: 120 -> 256 -> 768 -> 1024 -> 1369 -> 1369
  mlp_layer(zcat,  120,  256, 53, mA, false);
  mlp_layer(mA,    256,  768, 57, mB, false);
  mlp_layer(mB,    768, 1024, 61, mA, false);
  mlp_layer(mA,   1024, 1369, 65, mB, false);
  mlp_layer(mB,   1369, 1369, 69, xatt, true);
}